// TypeLayer_52896817218000
// MI455X (gfx1250) — compile-verified
//
#include <hip/hip_runtime.h>

typedef __attribute__((ext_vector_type(2))) float v2f;
typedef __attribute__((ext_vector_type(8))) float v8f;

#define HID 128  // D (hidden size), fixed by the reference

// ---------------------------------------------------------------------------
// Kernel 1: zero the entity accumulator (d_out), float4-wide.
// ---------------------------------------------------------------------------
__global__ void zero_f4(float4* __restrict__ p, int n4) {
  int i = blockIdx.x * blockDim.x + threadIdx.x;
  if (i < n4) p[i] = make_float4(0.f, 0.f, 0.f, 0.f);
}

// ---------------------------------------------------------------------------
// Kernel 2: rel_out[r,d] = sum_k rel_features[r,k] * W[d,k] + b[d]
// via V_WMMA_F32_16X16X4_F32. One wave32 per 16x16 output tile, K-loop of 32
// WMMA steps (K=128). Layouts per CDNA5 ISA 7.12.2:
//   A 16x4:  lane<16 -> M=lane,  {a.x,a.y} = K{kk,kk+1};  lane>=16 -> K{kk+2,kk+3}
//   B 4x16:  lane<16 -> N=lane,  {b.x,b.y} = K{kk,kk+1};  lane>=16 -> K{kk+2,kk+3}
//   C/D:     acc[v] = out[M = v + 8*(lane>=16)][N = lane&15]
// ---------------------------------------------------------------------------
__global__ void relout_gemm_wmma(const float* __restrict__ rel,
                                 const float* __restrict__ W,
                                 const float* __restrict__ bias,
                                 float* __restrict__ rel_out,
                                 int R) {
  const int colTiles = HID / 16;  // 8
  int wave = (blockIdx.x * blockDim.x + threadIdx.x) >> 5;
  int lane = threadIdx.x & 31;
  int rowTile = wave / colTiles;
  int colTile = wave - rowTile * colTiles;
  if (rowTile * 16 >= R) return;          // wave-uniform: EXEC stays all-1s

  int half = lane >> 4;                   // 0: lanes 0-15, 1: lanes 16-31
  int l    = lane & 15;
  int aRow = rowTile * 16 + l;
  if (aRow >= R) aRow = R - 1;            // clamp (R=2000 is a multiple of 16)
  int bCol = colTile * 16 + l;            // output feature d = column N

  const float* __restrict__ arow = rel + (size_t)aRow * HID;
  const float* __restrict__ brow = W   + (size_t)bCol * HID;

  v8f acc = {0.f, 0.f, 0.f, 0.f, 0.f, 0.f, 0.f, 0.f};
#pragma unroll 8
  for (int kk = 0; kk < HID; kk += 4) {
    int ka = kk + half * 2;
    v2f a; a.x = arow[ka]; a.y = arow[ka + 1];
    v2f b; b.x = brow[ka]; b.y = brow[ka + 1];
    // (neg_a, A, neg_b, B, c_mod, C, reuse_a, reuse_b)
    acc = __builtin_amdgcn_wmma_f32_16x16x4_f32(false, a, false, b,
                                                (short)0, acc, false, false);
  }

  float bv = bias[bCol];
  int rowBase = rowTile * 16 + half * 8;
  if (rowTile * 16 + 16 <= R) {
    // Full tile (the common case): wave-uniform branch, unguarded stores,
    // no per-element EXEC manipulation.
#pragma unroll
    for (int v = 0; v < 8; ++v) {
      rel_out[(size_t)(rowBase + v) * HID + bCol] = acc[v] + bv;
    }
  } else {
#pragma unroll
    for (int v = 0; v < 8; ++v) {
      int row = rowBase + v;
      if (row < R) rel_out[(size_t)row * HID + bCol] = acc[v] + bv;
    }
  }
}

// ---------------------------------------------------------------------------
// Kernel 3: per fact e: w = rel_out[rels[e]] * val[e];
//           out[tails[e]] += w; out[heads[e]] += w;   (f32 atomics)
// One wave per fact; each lane handles 4 consecutive features (float4 gather
// from the L2-resident 1MB rel_out table, 8 scalar atomic f32 adds).
// Index/value arrays are streamed once -> non-temporal loads so they do not
// evict the L2-resident accumulator (8MB) + rel_out table (1MB).
// ---------------------------------------------------------------------------
__global__ void scatter_facts(const int* __restrict__ rels,
                              const int* __restrict__ heads,
                              const int* __restrict__ tails,
                              const float* __restrict__ val,
                              const float* __restrict__ rel_out,
                              float* __restrict__ out, int E) {
  int gid  = blockIdx.x * blockDim.x + threadIdx.x;
  int e    = gid >> 5;
  int lane = gid & 31;
  if (e >= E) return;

  int r   = __builtin_nontemporal_load(rels + e);
  int h   = __builtin_nontemporal_load(heads + e);
  int t   = __builtin_nontemporal_load(tails + e);
  float v = __builtin_nontemporal_load(val + e);

  const float4* __restrict__ src = (const float4*)(rel_out + (size_t)r * HID);
  float4 w = src[lane];
  w.x *= v; w.y *= v; w.z *= v; w.w *= v;

  if (h == t) {  // self-loop: fold the two scatters into one
    float* o = out + (size_t)t * HID + lane * 4;
    atomicAdd(o + 0, 2.f * w.x); atomicAdd(o + 1, 2.f * w.y);
    atomicAdd(o + 2, 2.f * w.z); atomicAdd(o + 3, 2.f * w.w);
  } else {
    float* ot = out + (size_t)t * HID + lane * 4;
    float* oh = out + (size_t)h * HID + lane * 4;
    atomicAdd(ot + 0, w.x); atomicAdd(ot + 1, w.y);
    atomicAdd(ot + 2, w.z); atomicAdd(ot + 3, w.w);
    atomicAdd(oh + 0, w.x); atomicAdd(oh + 1, w.y);
    atomicAdd(oh + 2, w.z); atomicAdd(oh + 3, w.w);
  }
}

// ---------------------------------------------------------------------------
// Kernel 4: in-place ReLU on d_out, float4-wide.
// ---------------------------------------------------------------------------
__global__ void relu_f4(float4* __restrict__ p, int n4) {
  int i = blockIdx.x * blockDim.x + threadIdx.x;
  if (i < n4) {
    float4 x = p[i];
    x.x = fmaxf(x.x, 0.f); x.y = fmaxf(x.y, 0.f);
    x.z = fmaxf(x.z, 0.f); x.w = fmaxf(x.w, 0.f);
    p[i] = x;
  }
}

extern "C" void kernel_launch(void* const* d_in, const int* in_sizes, int n_in,
                              void* d_out, int out_size, void* d_ws, size_t ws_size,
                              hipStream_t stream) {
  // setup_inputs() order:
  // 0: local_entity [B,M] (unused — only shape matters)
  // 1: batch_heads [E]   2: batch_tails [E]   3: batch_rels [E]
  // 4: val_one [E]       5: rel_features [R,D]
  // 6: W [D,D]           7: b [D]
  const int*   heads = (const int*)d_in[1];
  const int*   tails = (const int*)d_in[2];
  const int*   rels  = (const int*)d_in[3];
  const float* val   = (const float*)d_in[4];
  const float* relf  = (const float*)d_in[5];
  const float* W     = (const float*)d_in[6];
  const float* bias  = (const float*)d_in[7];
  float*       out   = (float*)d_out;
  float*       rel_out = (float*)d_ws;   // R*HID floats (1 MB for R=2000)

  int E = in_sizes[1];
  int R = in_sizes[5] / HID;
  int n4 = out_size / 4;                 // out_size = B*M*HID floats

  // 1) zero accumulator
  zero_f4<<<(n4 + 255) / 256, 256, 0, stream>>>((float4*)out, n4);

  // 2) WMMA GEMM: rel_out = rel_features @ W.T + b
  int rtiles = (R + 15) / 16;
  int waves  = rtiles * (HID / 16);
  relout_gemm_wmma<<<(waves + 7) / 8, 256, 0, stream>>>(relf, W, bias, rel_out, R);

  // 3) gather + weighted scatter-add (one wave per fact)
  long long threads = (long long)E * 32;
  int sblocks = (int)((threads + 255) / 256);
  scatter_facts<<<sblocks, 256, 0, stream>>>(rels, heads, tails, val, rel_out, out, E);

  // 4) ReLU in place
  relu_f4<<<(n4 + 255) / 256, 256, 0, stream>>>((float4*)out, n4);
}